// BrainAGT_8435315769792
// MI455X (gfx1250) — compile-verified
//
#include <hip/hip_runtime.h>

#define B_  4
#define L_  1024
#define D_  512
#define H_  8
#define DK_ 64
#define M_  (B_*L_)   // 4096 rows

typedef __attribute__((ext_vector_type(2))) float v2f;
typedef __attribute__((ext_vector_type(8))) float v8f;

__device__ __forceinline__ v8f wmma4(v2f a, v2f b, v8f c) {
  // D = A(16x4 f32) * B(4x16 f32) + C(16x16 f32)
  return __builtin_amdgcn_wmma_f32_16x16x4_f32(
      /*neg_a=*/false, a, /*neg_b=*/false, b,
      /*c_mod=*/(short)0, c, /*reuse_a=*/false, /*reuse_b=*/false);
}

// Pipelined 32(M) x 64(N) x K wave GEMM core (2-stage register pipeline,
// peeled drain iteration so steady-state prefetch uses constant +8/+12
// element offsets -> immediate vmem offsets, minimal address ALU).
// a0r/a1r: per-lane A row pointers (rows m0+l16, m0+16+l16), already +2*half.
// bjr:     per-lane B row pointers (rows n0+j*16+l16 of [N,K] row-major),
//          already +2*half. K must be a multiple of 8, K >= 8.
// acc[s*4+j] = C subtile (M sub s, N sub j).
__device__ __forceinline__ void mm_pipe(
    const float* a0r, const float* a1r,
    const float* b0r, const float* b1r,
    const float* b2r, const float* b3r,
    int K, v8f acc[8])
{
  v2f A0s[2], A1s[2], Bs[2][4];
  A0s[0] = *(const v2f*)(a0r);       A1s[0] = *(const v2f*)(a1r);
  Bs[0][0] = *(const v2f*)(b0r);     Bs[0][1] = *(const v2f*)(b1r);
  Bs[0][2] = *(const v2f*)(b2r);     Bs[0][3] = *(const v2f*)(b3r);
  A0s[1] = *(const v2f*)(a0r + 4);   A1s[1] = *(const v2f*)(a1r + 4);
  Bs[1][0] = *(const v2f*)(b0r + 4); Bs[1][1] = *(const v2f*)(b1r + 4);
  Bs[1][2] = *(const v2f*)(b2r + 4); Bs[1][3] = *(const v2f*)(b3r + 4);

  #pragma unroll 1
  for (int k0 = 0; k0 < K - 8; k0 += 8) {
    {
      v2f a0 = A0s[0], a1 = A1s[0];
      v2f b0 = Bs[0][0], b1 = Bs[0][1], b2 = Bs[0][2], b3 = Bs[0][3];
      A0s[0] = *(const v2f*)(a0r + 8);  A1s[0] = *(const v2f*)(a1r + 8);
      Bs[0][0] = *(const v2f*)(b0r + 8); Bs[0][1] = *(const v2f*)(b1r + 8);
      Bs[0][2] = *(const v2f*)(b2r + 8); Bs[0][3] = *(const v2f*)(b3r + 8);
      acc[0] = wmma4(a0, b0, acc[0]); acc[1] = wmma4(a0, b1, acc[1]);
      acc[2] = wmma4(a0, b2, acc[2]); acc[3] = wmma4(a0, b3, acc[3]);
      acc[4] = wmma4(a1, b0, acc[4]); acc[5] = wmma4(a1, b1, acc[5]);
      acc[6] = wmma4(a1, b2, acc[6]); acc[7] = wmma4(a1, b3, acc[7]);
    }
    {
      v2f a0 = A0s[1], a1 = A1s[1];
      v2f b0 = Bs[1][0], b1 = Bs[1][1], b2 = Bs[1][2], b3 = Bs[1][3];
      A0s[1] = *(const v2f*)(a0r + 12);  A1s[1] = *(const v2f*)(a1r + 12);
      Bs[1][0] = *(const v2f*)(b0r + 12); Bs[1][1] = *(const v2f*)(b1r + 12);
      Bs[1][2] = *(const v2f*)(b2r + 12); Bs[1][3] = *(const v2f*)(b3r + 12);
      acc[0] = wmma4(a0, b0, acc[0]); acc[1] = wmma4(a0, b1, acc[1]);
      acc[2] = wmma4(a0, b2, acc[2]); acc[3] = wmma4(a0, b3, acc[3]);
      acc[4] = wmma4(a1, b0, acc[4]); acc[5] = wmma4(a1, b1, acc[5]);
      acc[6] = wmma4(a1, b2, acc[6]); acc[7] = wmma4(a1, b3, acc[7]);
    }
    a0r += 8; a1r += 8; b0r += 8; b1r += 8; b2r += 8; b3r += 8;
  }
  // drain: consume the two in-flight stages, no more prefetch
  {
    v2f a0 = A0s[0], a1 = A1s[0];
    v2f b0 = Bs[0][0], b1 = Bs[0][1], b2 = Bs[0][2], b3 = Bs[0][3];
    acc[0] = wmma4(a0, b0, acc[0]); acc[1] = wmma4(a0, b1, acc[1]);
    acc[2] = wmma4(a0, b2, acc[2]); acc[3] = wmma4(a0, b3, acc[3]);
    acc[4] = wmma4(a1, b0, acc[4]); acc[5] = wmma4(a1, b1, acc[5]);
    acc[6] = wmma4(a1, b2, acc[6]); acc[7] = wmma4(a1, b3, acc[7]);
  }
  {
    v2f a0 = A0s[1], a1 = A1s[1];
    v2f b0 = Bs[1][0], b1 = Bs[1][1], b2 = Bs[1][2], b3 = Bs[1][3];
    acc[0] = wmma4(a0, b0, acc[0]); acc[1] = wmma4(a0, b1, acc[1]);
    acc[2] = wmma4(a0, b2, acc[2]); acc[3] = wmma4(a0, b3, acc[3]);
    acc[4] = wmma4(a1, b0, acc[4]); acc[5] = wmma4(a1, b1, acc[5]);
    acc[6] = wmma4(a1, b2, acc[6]); acc[7] = wmma4(a1, b3, acc[7]);
  }
}

// ---------------- block reduction helpers (wave32, 8 waves/block) ----------
__device__ __forceinline__ float blk_sum(float val, float* red) {
  #pragma unroll
  for (int o = 16; o; o >>= 1) val += __shfl_xor(val, o, 32);
  __syncthreads();
  if ((threadIdx.x & 31) == 0) red[threadIdx.x >> 5] = val;
  __syncthreads();
  float s = 0.0f;
  #pragma unroll
  for (int i = 0; i < 8; ++i) s += red[i];
  return s;
}

__device__ __forceinline__ float blk_max(float val, float* red) {
  #pragma unroll
  for (int o = 16; o; o >>= 1) val = fmaxf(val, __shfl_xor(val, o, 32));
  __syncthreads();
  if ((threadIdx.x & 31) == 0) red[threadIdx.x >> 5] = val;
  __syncthreads();
  float s = red[0];
  #pragma unroll
  for (int i = 1; i < 8; ++i) s = fmaxf(s, red[i]);
  return s;
}

// ---------------- GEMM: out = A[M,K] @ W[N,K]^T + bias[N] ------------------
// out_mode 0: out[m*N + n]                         (row-major [M,N])
// out_mode 1: out[((b*H+h)*L + l)*DK + d]          (split heads [B,H,L,dk])
// out_mode 2: out[((b*H+h)*DK + d)*L + l]          (split heads, transposed [B,H,dk,L])
__global__ __launch_bounds__(256) void k_gemm_xwt(
    const float* __restrict__ A, const float* __restrict__ W,
    const float* __restrict__ bias, float* __restrict__ out,
    int K, int N, int out_mode)
{
  const int lane = threadIdx.x & 31;
  const int wave = threadIdx.x >> 5;
  const int half = lane >> 4;
  const int l16  = lane & 15;
  const int m0 = blockIdx.x * 256 + wave * 32;
  const int n0 = blockIdx.y * 64;

  const float* a0r = A + (size_t)(m0 + l16) * K + 2 * half;
  const float* a1r = A + (size_t)(m0 + 16 + l16) * K + 2 * half;
  const float* b0r = W + (size_t)(n0 +  0 + l16) * K + 2 * half;
  const float* b1r = W + (size_t)(n0 + 16 + l16) * K + 2 * half;
  const float* b2r = W + (size_t)(n0 + 32 + l16) * K + 2 * half;
  const float* b3r = W + (size_t)(n0 + 48 + l16) * K + 2 * half;

  v8f acc[8] = { {}, {}, {}, {}, {}, {}, {}, {} };
  mm_pipe(a0r, a1r, b0r, b1r, b2r, b3r, K, acc);

  #pragma unroll
  for (int s = 0; s < 2; ++s) {
    #pragma unroll
    for (int j = 0; j < 4; ++j) {
      const int n = n0 + j * 16 + l16;
      const float bv = bias[n];
      #pragma unroll
      for (int r = 0; r < 8; ++r) {
        const int m = m0 + s * 16 + r + 8 * half;
        const float val = acc[s * 4 + j][r] + bv;
        if (out_mode == 0) {
          out[(size_t)m * N + n] = val;
        } else {
          const int b = m >> 10, l = m & (L_ - 1);
          const int h = n >> 6,  d = n & (DK_ - 1);
          if (out_mode == 1)
            out[(((size_t)b * H_ + h) * L_ + l) * DK_ + d] = val;
          else
            out[(((size_t)b * H_ + h) * DK_ + d) * L_ + l] = val;
        }
      }
    }
  }
}

// ---------------- row norms ||q||^2, ||k||^2 + zero dist_max ---------------
__global__ __launch_bounds__(256) void k_norms(
    const float* __restrict__ q, const float* __restrict__ k,
    float* __restrict__ qq, float* __restrict__ kk, float* __restrict__ dmax)
{
  const int idx = blockIdx.x * 256 + threadIdx.x;
  if (idx < B_ * H_) dmax[idx] = 0.0f;      // reset every call
  const int rows = B_ * H_ * L_;            // 32768
  if (idx >= 2 * rows) return;
  const float* src = (idx < rows) ? q : k;
  float* dst = (idx < rows) ? qq : kk;
  const int r = idx & (rows - 1);
  const float* p = src + (size_t)r * DK_;
  float s = 0.0f;
  for (int d = 0; d < DK_; ++d) s += p[d] * p[d];
  dst[r] = s;
}

// ---------------- S = q k^T; dist = max(qq+kk-2S,0)/dk; track max ----------
__global__ __launch_bounds__(256) void k_qk_dist(
    const float* __restrict__ q, const float* __restrict__ k,
    const float* __restrict__ qq, const float* __restrict__ kk,
    float* __restrict__ dist, float* __restrict__ dmax)
{
  const int lane = threadIdx.x & 31;
  const int wave = threadIdx.x >> 5;
  const int half = lane >> 4;
  const int l16  = lane & 15;
  const int z  = blockIdx.z;                    // b*H + h
  const int m0 = blockIdx.x * 256 + wave * 32;
  const int n0 = blockIdx.y * 64;
  const float* Q  = q + (size_t)z * L_ * DK_;
  const float* Km = k + (size_t)z * L_ * DK_;

  const float* a0r = Q  + (size_t)(m0 + l16) * DK_ + 2 * half;
  const float* a1r = Q  + (size_t)(m0 + 16 + l16) * DK_ + 2 * half;
  const float* b0r = Km + (size_t)(n0 +  0 + l16) * DK_ + 2 * half;
  const float* b1r = Km + (size_t)(n0 + 16 + l16) * DK_ + 2 * half;
  const float* b2r = Km + (size_t)(n0 + 32 + l16) * DK_ + 2 * half;
  const float* b3r = Km + (size_t)(n0 + 48 + l16) * DK_ + 2 * half;

  v8f acc[8] = { {}, {}, {}, {}, {}, {}, {}, {} };
  mm_pipe(a0r, a1r, b0r, b1r, b2r, b3r, DK_, acc);

  float lmax = 0.0f;
  float* drow = dist + (size_t)z * L_ * L_;
  #pragma unroll
  for (int s = 0; s < 2; ++s) {
    #pragma unroll
    for (int j = 0; j < 4; ++j) {
      const int n = n0 + j * 16 + l16;
      const float kkn = kk[z * L_ + n];
      #pragma unroll
      for (int r = 0; r < 8; ++r) {
        const int m = m0 + s * 16 + r + 8 * half;
        float d2 = qq[z * L_ + m] + kkn - 2.0f * acc[s * 4 + j][r];
        d2 = fmaxf(d2, 0.0f) * (1.0f / (float)DK_);
        drow[(size_t)m * L_ + n] = d2;
        lmax = fmaxf(lmax, d2);
      }
    }
  }
  #pragma unroll
  for (int o = 16; o; o >>= 1) lmax = fmaxf(lmax, __shfl_xor(lmax, o, 32));
  if (lane == 0)  // values >= 0, so int-compare == float-compare
    atomicMax((int*)(dmax + z), __float_as_int(lmax));
}

// ---------------- per-row sparsemax over logits (in-place) -----------------
__global__ __launch_bounds__(256) void k_sparsemax(
    float* __restrict__ attn, const float* __restrict__ dmax,
    const float* __restrict__ theta)
{
  __shared__ float red[8];
  const int row = blockIdx.x;           // bh*L + i
  const int bh  = row >> 10;
  const int h   = bh & (H_ - 1);
  const float coef = expf(theta[h]);
  const float inv  = 1.0f / (dmax[bh] + 1e-6f);
  float* p = attn + (size_t)row * L_;
  const int tid = threadIdx.x;

  float x[4];
  #pragma unroll
  for (int t = 0; t < 4; ++t)
    x[t] = expf(-coef * p[tid + 256 * t] * inv);

  // sparsemax is shift-invariant -> skip the reference's (x - max) shift.
  float vm = fmaxf(fmaxf(x[0], x[1]), fmaxf(x[2], x[3]));
  vm = blk_max(vm, red);

  // f(tau) = sum relu(x - tau) is piecewise-linear decreasing; root in [vm-1, vm]
  float lo = vm - 1.0f, hi = vm;
  for (int it = 0; it < 32; ++it) {
    const float tau = 0.5f * (lo + hi);
    float s = 0.0f;
    #pragma unroll
    for (int t = 0; t < 4; ++t) s += fmaxf(x[t] - tau, 0.0f);
    s = blk_sum(s, red);
    if (s >= 1.0f) lo = tau; else hi = tau;
  }
  const float tau = 0.5f * (lo + hi);
  float cnt = 0.0f, s = 0.0f;
  #pragma unroll
  for (int t = 0; t < 4; ++t)
    if (x[t] > tau) { cnt += 1.0f; s += x[t]; }
  cnt = blk_sum(cnt, red);
  s   = blk_sum(s, red);
  const float tf = (s - 1.0f) / fmaxf(cnt, 1.0f);   // exact tau given support
  #pragma unroll
  for (int t = 0; t < 4; ++t)
    p[tid + 256 * t] = fmaxf(x[t] - tf, 0.0f);
}

// ---------------- context = attn @ v  (v stored transposed [B,H,dk,L]) -----
__global__ __launch_bounds__(256) void k_attn_v(
    const float* __restrict__ attn, const float* __restrict__ vt,
    float* __restrict__ ctx)
{
  const int lane = threadIdx.x & 31;
  const int wave = threadIdx.x >> 5;
  const int half = lane >> 4;
  const int l16  = lane & 15;
  const int z  = blockIdx.z;                 // b*H + h
  const int m0 = blockIdx.x * 256 + wave * 32;
  const float* A = attn + (size_t)z * L_ * L_;
  const float* V = vt   + (size_t)z * DK_ * L_;   // [dk, L] row-major == [N,K]

  const float* a0r = A + (size_t)(m0 + l16) * L_ + 2 * half;
  const float* a1r = A + (size_t)(m0 + 16 + l16) * L_ + 2 * half;
  const float* b0r = V + (size_t)( 0 + l16) * L_ + 2 * half;
  const float* b1r = V + (size_t)(16 + l16) * L_ + 2 * half;
  const float* b2r = V + (size_t)(32 + l16) * L_ + 2 * half;
  const float* b3r = V + (size_t)(48 + l16) * L_ + 2 * half;

  v8f acc[8] = { {}, {}, {}, {}, {}, {}, {}, {} };
  mm_pipe(a0r, a1r, b0r, b1r, b2r, b3r, L_, acc);

  const int b = z >> 3, h = z & (H_ - 1);
  #pragma unroll
  for (int s = 0; s < 2; ++s) {
    #pragma unroll
    for (int j = 0; j < 4; ++j) {
      const int n = j * 16 + l16;
      #pragma unroll
      for (int r = 0; r < 8; ++r) {
        const int m = m0 + s * 16 + r + 8 * half;
        ctx[((size_t)(b * L_ + m)) * D_ + h * DK_ + n] = acc[s * 4 + j][r];
      }
    }
  }
}

// ---------------- mean over heads -> output part 2 -------------------------
__global__ __launch_bounds__(256) void k_mean(
    const float* __restrict__ attn, float* __restrict__ out2)
{
  const size_t idx = (size_t)blockIdx.x * 256 + threadIdx.x;
  if (idx >= (size_t)B_ * L_ * L_) return;
  const size_t b   = idx >> 20;              // / (L*L)
  const size_t rem = idx & ((1u << 20) - 1);
  float s = 0.0f;
  #pragma unroll
  for (int h = 0; h < H_; ++h)
    s += attn[(((size_t)b * H_ + h) << 20) + rem];
  out2[idx] = s * (1.0f / (float)H_);
}

extern "C" void kernel_launch(void* const* d_in, const int* in_sizes, int n_in,
                              void* d_out, int out_size, void* d_ws, size_t ws_size,
                              hipStream_t stream) {
  (void)in_sizes; (void)n_in; (void)out_size; (void)ws_size;
  const float* x  = (const float*)d_in[0];
  const float* Wq = (const float*)d_in[1];
  const float* bq = (const float*)d_in[2];
  const float* Wk = (const float*)d_in[3];
  const float* bk = (const float*)d_in[4];
  const float* Wv = (const float*)d_in[5];
  const float* bv = (const float*)d_in[6];
  const float* Wo = (const float*)d_in[7];
  const float* bo = (const float*)d_in[8];
  const float* th = (const float*)d_in[9];
  float* out = (float*)d_out;

  // workspace layout (floats); total ~42.0M floats (~168 MB)
  float* ws   = (float*)d_ws;
  float* dist = ws;                          // B*H*L*L = 33554432 (attn in-place)
  float* q    = ws + 33554432;               // 2097152
  float* k    = ws + 35651584;               // 2097152
  float* vt   = ws + 37748736;               // 2097152 ([B,H,dk,L])
  float* ctx  = ws + 39845888;               // 2097152
  float* qq   = ws + 41943040;               // 32768
  float* kk   = ws + 41975808;               // 32768
  float* dmax = ws + 42008576;               // 32

  const dim3 blk(256);
  const dim3 gProj(M_ / 256, D_ / 64);       // 16 x 8

  k_gemm_xwt<<<gProj, blk, 0, stream>>>(x, Wq, bq, q,  D_, D_, 1);
  k_gemm_xwt<<<gProj, blk, 0, stream>>>(x, Wk, bk, k,  D_, D_, 1);
  k_gemm_xwt<<<gProj, blk, 0, stream>>>(x, Wv, bv, vt, D_, D_, 2);

  k_norms<<<dim3(256), blk, 0, stream>>>(q, k, qq, kk, dmax);

  k_qk_dist<<<dim3(L_ / 256, L_ / 64, B_ * H_), blk, 0, stream>>>(
      q, k, qq, kk, dist, dmax);

  k_sparsemax<<<dim3(B_ * H_ * L_), blk, 0, stream>>>(dist, dmax, th);

  k_attn_v<<<dim3(L_ / 256, 1, B_ * H_), blk, 0, stream>>>(dist, vt, ctx);

  k_mean<<<dim3((B_ * L_ * L_) / 256), blk, 0, stream>>>(
      dist, out + (size_t)M_ * D_);

  k_gemm_xwt<<<gProj, blk, 0, stream>>>(ctx, Wo, bo, out, D_, D_, 0);
}